// HistoryCompression_69157563400234
// MI455X (gfx1250) — compile-verified
//
#include <hip/hip_runtime.h>

typedef __attribute__((ext_vector_type(2))) float v2f;
typedef __attribute__((ext_vector_type(8))) float v8f;
typedef __attribute__((ext_vector_type(4))) int   v4i;

#define S_LEN 8192
#define D_DIM 2048
#define W_STRIDE (2 * D_DIM)

// ---- CDNA5 async global->LDS path (guarded so compile never regresses) ----
#if __has_builtin(__builtin_amdgcn_global_load_async_to_lds_b128) && \
    __has_builtin(__builtin_amdgcn_s_wait_asynccnt)
#define HAVE_ASYNC_LDS 1
#else
#define HAVE_ASYNC_LDS 0
#endif

#if HAVE_ASYNC_LDS
typedef __attribute__((address_space(1))) v4i* gbl_v4i_ptr;
typedef __attribute__((address_space(3))) v4i* lds_v4i_ptr;
__device__ __forceinline__ void async_cp16(void* lds_dst, const void* gsrc) {
    __builtin_amdgcn_global_load_async_to_lds_b128(
        (gbl_v4i_ptr)(void*)gsrc, (lds_v4i_ptr)lds_dst, /*offset*/0, /*cpol*/0);
}
#endif

// ---------------------------------------------------------------------------
// Kernel 0: init state (-1.0) and grid-barrier counter (deterministic per call)
// ---------------------------------------------------------------------------
__global__ void hcp_init_kernel(float* stateA, unsigned* counter) {
    int i = blockIdx.x * blockDim.x + threadIdx.x;
    if (i < D_DIM) stateA[i] = -1.0f;
    if (i == 0) *counter = 0u;
}

// ---------------------------------------------------------------------------
// Kernel 1: column-wise cummax over S (streaming, coalesced across columns)
// ---------------------------------------------------------------------------
__global__ void hcp_cummax_kernel(const float* __restrict__ x,
                                  float* __restrict__ hc) {
    int col = blockIdx.x * blockDim.x + threadIdx.x;
    if (col >= D_DIM) return;
    float m = -__builtin_inff();
    for (int t = 0; t < S_LEN; ++t) {
        if (t + 16 < S_LEN)
            __builtin_prefetch(&x[(size_t)(t + 16) * D_DIM + col], 0, 0);
        float v = x[(size_t)t * D_DIM + col];
        m = fmaxf(m, v);
        hc[(size_t)t * D_DIM + col] = m;
    }
}

// ---------------------------------------------------------------------------
// Kernel 2: ctx_proj = context_enc @ W_ctx.T + b  via V_WMMA_F32_16X16X4_F32
//   Block tile 32(M) x 128(N), K-block 64. 8 waves; each wave owns a 16x32
//   tile = two independent WMMA accumulation chains (ILP in the matrix pipe).
//   LDS tiles double-buffered and filled with GLOBAL_LOAD_ASYNC_TO_LDS_B128
//   so the next tile streams in under the current tile's 32 WMMAs.
// ---------------------------------------------------------------------------
#define BM 32
#define BN 128
#define BK 64
#define LDS_STRIDE 68                         // rows 16B aligned, bank-skewed
#define TILE_FLOATS ((BM + BN) * LDS_STRIDE)  // one buffer: As + Bs

__global__ __launch_bounds__(256) void hcp_ctxproj_wmma_kernel(
    const float* __restrict__ A,    // (S, D) context_enc
    const float* __restrict__ W,    // (D, 2D)
    const float* __restrict__ bias, // (D)
    float* __restrict__ C)          // (S, D) ctx_proj (staged in d_out)
{
    extern __shared__ float lds[];  // 2 * TILE_FLOATS

    const int tid  = threadIdx.x;
    const int m0   = blockIdx.y * BM;
    const int n0   = blockIdx.x * BN;
    const int wave = tid >> 5;      // 0..7
    const int lane = tid & 31;
    const int half = lane >> 4;     // 0: lanes 0-15, 1: lanes 16-31
    const int lm   = lane & 15;
    const int wm   = wave >> 2;     // 0..1 : M sub-tile
    const int wn   = wave & 3;      // 0..3 : N sub-tile (32 cols each)

    // per-thread tile-fill coordinates (float4 granular)
    // A: 32x64 = 512 float4 -> 2/thread ; B: 128x64 = 2048 float4 -> 8/thread
    v8f acc0 = {};
    v8f acc1 = {};

#if HAVE_ASYNC_LDS
    // ---- async double-buffered pipeline ----
    {   // prologue: fill buffer 0 with k-block 0
        float* As = lds;
        float* Bs = lds + BM * LDS_STRIDE;
        #pragma unroll
        for (int q = 0; q < 2; ++q) {
            int v = tid + q * 256, r = v >> 4, c4 = (v & 15) << 2;
            async_cp16(&As[r * LDS_STRIDE + c4],
                       A + (size_t)(m0 + r) * D_DIM + c4);
        }
        #pragma unroll
        for (int q = 0; q < 8; ++q) {
            int v = tid + q * 256, n = v >> 4, c4 = (v & 15) << 2;
            async_cp16(&Bs[n * LDS_STRIDE + c4],
                       W + (size_t)(n0 + n) * W_STRIDE + D_DIM + c4);
        }
    }
#endif

    for (int kb = 0; kb < D_DIM; kb += BK) {
#if HAVE_ASYNC_LDS
        float* cur = lds + (((unsigned)kb >> 6) & 1u) * TILE_FLOATS;
        float* nxt = lds + ((((unsigned)kb >> 6) + 1u) & 1u) * TILE_FLOATS;
        __builtin_amdgcn_s_wait_asynccnt(0);   // current tile resident
        __syncthreads();                       // everyone past prior compute
        if (kb + BK < D_DIM) {                 // stream next tile under WMMAs
            float* As = nxt;
            float* Bs = nxt + BM * LDS_STRIDE;
            int kn = kb + BK;
            #pragma unroll
            for (int q = 0; q < 2; ++q) {
                int v = tid + q * 256, r = v >> 4, c4 = (v & 15) << 2;
                async_cp16(&As[r * LDS_STRIDE + c4],
                           A + (size_t)(m0 + r) * D_DIM + kn + c4);
            }
            #pragma unroll
            for (int q = 0; q < 8; ++q) {
                int v = tid + q * 256, n = v >> 4, c4 = (v & 15) << 2;
                async_cp16(&Bs[n * LDS_STRIDE + c4],
                           W + (size_t)(n0 + n) * W_STRIDE + D_DIM + kn + c4);
            }
        }
#else
        float* cur = lds;
        {   // synchronous tile fill (fallback)
            float* As = cur;
            float* Bs = cur + BM * LDS_STRIDE;
            #pragma unroll
            for (int q = 0; q < 2; ++q) {
                int v = tid + q * 256, r = v >> 4, c4 = (v & 15) << 2;
                float4 s = *(const float4*)(A + (size_t)(m0 + r) * D_DIM + kb + c4);
                *(float4*)(&As[r * LDS_STRIDE + c4]) = s;
            }
            #pragma unroll
            for (int q = 0; q < 8; ++q) {
                int v = tid + q * 256, n = v >> 4, c4 = (v & 15) << 2;
                float4 s = *(const float4*)(W + (size_t)(n0 + n) * W_STRIDE + D_DIM + kb + c4);
                *(float4*)(&Bs[n * LDS_STRIDE + c4]) = s;
            }
        }
        __syncthreads();
#endif

        const float* aRow  = cur + (wm * 16 + lm) * LDS_STRIDE;
        const float* b0Row = cur + BM * LDS_STRIDE + (wn * 32 + lm) * LDS_STRIDE;
        const float* b1Row = b0Row + 16 * LDS_STRIDE;
        #pragma unroll
        for (int kk = 0; kk < BK; kk += 4) {
            // A 16x4: lanes0-15 K={kk,kk+1}, lanes16-31 K={kk+2,kk+3}
            v2f a  = *(const v2f*)(aRow + kk + 2 * half);
            // B 4x16: B[k][n] = Bs[n][k] -> contiguous pair per lane
            v2f b0 = *(const v2f*)(b0Row + kk + 2 * half);
            v2f b1 = *(const v2f*)(b1Row + kk + 2 * half);
            acc0 = __builtin_amdgcn_wmma_f32_16x16x4_f32(
                false, a, false, b0, (short)0, acc0, false, false);
            acc1 = __builtin_amdgcn_wmma_f32_16x16x4_f32(
                false, a, false, b1, (short)0, acc1, false, false);
        }
#if !HAVE_ASYNC_LDS
        __syncthreads();
#endif
    }

    // C/D layout: VGPR v -> M = v + 8*half, N = lane%16
    const int n_a = n0 + wn * 32 + lm;
    const int n_b = n_a + 16;
    const float bv_a = bias[n_a];
    const float bv_b = bias[n_b];
    #pragma unroll
    for (int v = 0; v < 8; ++v) {
        int m = m0 + wm * 16 + v + 8 * half;
        C[(size_t)m * D_DIM + n_a] = acc0[v] + bv_a;
        C[(size_t)m * D_DIM + n_b] = acc1[v] + bv_b;
    }
}

// ---------------------------------------------------------------------------
// Kernel 3: sequential scan. Persistent grid of 64 blocks; each block pins its
// 32-row slice of W_state (256KB) in the 320KB LDS. Per step: async-stage the
// 8KB state vector into LDS, 8-threads-per-row GEMV from LDS, sigmoid gate,
// write state slice, device-scope grid barrier.
// ---------------------------------------------------------------------------
#define NBLK 64
#define ROWS_PER_BLK (D_DIM / NBLK)   // 32
#define WL_STRIDE 2050                // even (8B-aligned b64) + bank skew of 2

__global__ __launch_bounds__(256) void hcp_scan_kernel(
    const float* __restrict__ W,   // (D, 2D): W_state = cols [0, D)
    const float* __restrict__ hc,  // (S, D)
    float* out,                    // holds ctx_proj; overwritten with history
    float* stateA, float* stateB,
    unsigned* counter)
{
    extern __shared__ float lds[];
    float* Wl      = lds;                                  // 32 * 2050
    float* statel  = lds + ROWS_PER_BLK * WL_STRIDE;       // 2048 (16B aligned)
    float* partial = statel + D_DIM;                       // 256

    const int tid = threadIdx.x;
    const int i0  = blockIdx.x * ROWS_PER_BLK;

    // One-time: pin W_state slice in LDS (coalesced global reads)
    for (int idx = tid; idx < ROWS_PER_BLK * D_DIM; idx += 256) {
        int r = idx >> 11;
        int j = idx & (D_DIM - 1);
        Wl[r * WL_STRIDE + j] = W[(size_t)(i0 + r) * W_STRIDE + j];
    }

    const int r = tid >> 3;        // output row within slice: 0..31
    const int c = tid & 7;         // j-chunk: 0..7  (256 elems each)
    const int jbase = c * 256;

    const float* stateCur = stateA;
    float*       stateNxt = stateB;

    for (int t = 0; t < S_LEN; ++t) {
        // Stage state vector (8KB) into LDS — on the serial critical path.
#if HAVE_ASYNC_LDS
        #pragma unroll
        for (int q = 0; q < 2; ++q) {
            int idx = (tid + q * 256) << 2;    // float4 id * 4
            async_cp16(&statel[idx], &stateCur[idx]);
        }
        __builtin_amdgcn_s_wait_asynccnt(0);
#else
        #pragma unroll
        for (int q = 0; q < 4; ++q) {
            int j = (tid + q * 256) * 2;
            *(float2*)(&statel[j]) = *(const float2*)(&stateCur[j]);
        }
#endif
        __syncthreads();

        // Partial dot: 256 MACs from LDS, staggered start to spread banks
        float sum = 0.0f;
        #pragma unroll 4
        for (int jj = 0; jj < 256; jj += 2) {
            int j = jbase + ((jj + 8 * c) & 255);
            v2f w = *(const v2f*)(&Wl[r * WL_STRIDE + j]);
            v2f s = *(const v2f*)(&statel[j]);
            sum = fmaf(w.x, s.x, sum);
            sum = fmaf(w.y, s.y, sum);
        }
        partial[tid] = sum;
        __syncthreads();

        if (tid < ROWS_PER_BLK) {
            int i = i0 + tid;
            float acc = 0.0f;
            #pragma unroll
            for (int q = 0; q < 8; ++q) acc += partial[tid * 8 + q];
            float pre = acc + out[(size_t)t * D_DIM + i];  // ctx_proj staged
            float g   = 1.0f / (1.0f + __expf(-pre));
            float ns  = hc[(size_t)t * D_DIM + i] * g;
            stateNxt[i] = ns;
            out[(size_t)t * D_DIM + i] = ns;               // reset_history
            if (t == S_LEN - 1) out[(size_t)S_LEN * D_DIM + i] = ns; // final
        }
        __threadfence();
        __syncthreads();

        // Device-scope grid barrier (one arrival per block)
        if (tid == 0) {
            __hip_atomic_fetch_add(counter, 1u, __ATOMIC_ACQ_REL,
                                   __HIP_MEMORY_SCOPE_AGENT);
            unsigned target = (unsigned)(t + 1) * NBLK;
            while (__hip_atomic_load(counter, __ATOMIC_ACQUIRE,
                                     __HIP_MEMORY_SCOPE_AGENT) < target) {
                __builtin_amdgcn_s_sleep(1);
            }
        }
        __syncthreads();

        // swap double buffers
        float* tmp = (float*)stateCur;
        stateCur = stateNxt;
        stateNxt = tmp;
    }
}

// ---------------------------------------------------------------------------
extern "C" void kernel_launch(void* const* d_in, const int* in_sizes, int n_in,
                              void* d_out, int out_size, void* d_ws, size_t ws_size,
                              hipStream_t stream) {
    const float* x  = (const float*)d_in[0];   // (S, D)
    const float* W  = (const float*)d_in[1];   // (D, 2D)
    const float* b  = (const float*)d_in[2];   // (D)
    float* out = (float*)d_out;                // S*D history + D final
    float* ws  = (float*)d_ws;

    float*    hc      = ws;                                  // S*D floats
    float*    stateA  = ws + (size_t)S_LEN * D_DIM;          // D
    float*    stateB  = stateA + D_DIM;                      // D
    unsigned* counter = (unsigned*)(stateB + D_DIM);

    const size_t gemm_shmem = (size_t)(2 * TILE_FLOATS) * sizeof(float);
    const size_t scan_shmem =
        (size_t)(ROWS_PER_BLK * WL_STRIDE + D_DIM + 256) * sizeof(float);
    (void)hipFuncSetAttribute((const void*)hcp_ctxproj_wmma_kernel,
                              hipFuncAttributeMaxDynamicSharedMemorySize,
                              (int)gemm_shmem);
    (void)hipFuncSetAttribute((const void*)hcp_scan_kernel,
                              hipFuncAttributeMaxDynamicSharedMemorySize,
                              (int)scan_shmem);

    hcp_init_kernel<<<8, 256, 0, stream>>>(stateA, counter);
    hcp_cummax_kernel<<<D_DIM / 256, 256, 0, stream>>>(x, hc);

    dim3 gemm_grid(D_DIM / BN, S_LEN / BM);
    hcp_ctxproj_wmma_kernel<<<gemm_grid, 256, gemm_shmem, stream>>>(x, W, b, out);

    hcp_scan_kernel<<<NBLK, 256, scan_shmem, stream>>>(W, hc, out, stateA,
                                                       stateB, counter);
}